// GCN_27084063769011
// MI455X (gfx1250) — compile-verified
//
#include <hip/hip_runtime.h>

#define N_NODES 50000
#define E_EDGES 800000
#define F_IN    128
#define F_HID   128
#define F_OUT   64

typedef __attribute__((ext_vector_type(2))) float v2f;
typedef __attribute__((ext_vector_type(8))) float v8f;

// ---------------------------------------------------------------------------
// Init: deg = 1.0 (self loops), zero layer-1 accumulator and d_out.
// ---------------------------------------------------------------------------
__global__ void k_init(float* __restrict__ deg, float* __restrict__ agg1,
                       float* __restrict__ out) {
    int i = blockIdx.x * blockDim.x + threadIdx.x;
    if (i < N_NODES * F_HID) agg1[i] = 0.0f;
    if (i < N_NODES * F_OUT) out[i]  = 0.0f;
    if (i < N_NODES)         deg[i]  = 1.0f;   // self-loop contribution
}

// ---------------------------------------------------------------------------
// Degree histogram over dst (L2-resident float atomics).
// ---------------------------------------------------------------------------
__global__ void k_degree(const long long* __restrict__ dst, float* __restrict__ deg) {
    int e = blockIdx.x * blockDim.x + threadIdx.x;
    if (e < E_EDGES) atomicAdd(&deg[(int)dst[e]], 1.0f);
}

__global__ void k_rsqrt(float* __restrict__ deg) {
    int i = blockIdx.x * blockDim.x + threadIdx.x;
    if (i < N_NODES) deg[i] = rsqrtf(deg[i]);  // deg -> dinv in place
}

// ---------------------------------------------------------------------------
// Transpose weights once so the WMMA B fragment is a contiguous float2 load.
// in: [R,C] row-major -> out: [C,R]
// ---------------------------------------------------------------------------
template<int R, int C>
__global__ void k_transpose(const float* __restrict__ in, float* __restrict__ out) {
    int i = blockIdx.x * blockDim.x + threadIdx.x;
    if (i < R * C) {
        int r = i / C, c = i - r * C;
        out[(size_t)c * R + r] = in[i];
    }
}

// ---------------------------------------------------------------------------
// FP32 WMMA GEMM: C[M,NOUT] = A[M,K] * Bt^T, with Bt = B transposed [NOUT,K].
// One wave per 16x16 C tile; K-loop in steps of 4 via V_WMMA_F32_16X16X4_F32.
// Both A and B fragments are contiguous float2 (global_load_b64) per step.
// blockDim.x = (NOUT/16)*32; gridDim.x = M/16.
// ---------------------------------------------------------------------------
template<int K, int NOUT>
__global__ void k_gemm_wmma(const float* __restrict__ A,
                            const float* __restrict__ Bt,   // [NOUT, K]
                            float* __restrict__ C) {
    const int lane = threadIdx.x & 31;
    const int wave = threadIdx.x >> 5;
    const int m0   = blockIdx.x * 16;
    const int n0   = wave * 16;
    const int hi   = lane >> 4;    // 0: K pair {0,1}/rows 0-7 of C; 1: {2,3}/rows 8-15
    const int lr   = lane & 15;

    const float* arow = A  + (size_t)(m0 + lr) * K + 2 * hi;
    const float* brow = Bt + (size_t)(n0 + lr) * K + 2 * hi;
    v8f acc = {};

    for (int k = 0; k < K; k += 4) {
        // A fragment (16x4 f32): lane lr holds A[m0+lr][kk..kk+1] (contiguous)
        const float2 ap = *(const float2*)(arow + k);
        // B fragment (4x16): lane lr holds B[kk..kk+1][n0+lr] = Bt[n0+lr][kk..kk+1]
        const float2 bp = *(const float2*)(brow + k);
        v2f a; a.x = ap.x; a.y = ap.y;
        v2f b; b.x = bp.x; b.y = bp.y;
        acc = __builtin_amdgcn_wmma_f32_16x16x4_f32(
            /*neg_a=*/false, a, /*neg_b=*/false, b,
            /*c_mod=*/(short)0, acc, /*reuse_a=*/false, /*reuse_b=*/false);
    }

    // C/D layout: VGPR i -> row m0 + 8*hi + i, column n0 + lr
#pragma unroll
    for (int i = 0; i < 8; ++i) {
        C[(size_t)(m0 + 8 * hi + i) * NOUT + n0 + lr] = acc[i];
    }
}

// ---------------------------------------------------------------------------
// Edge aggregation: out[dst] += h[src] * dinv[src]*dinv[dst].
// One wave per logical edge (E_EDGES real edges + N_NODES self loops).
// src/dst/norm are wave-uniform -> force into SGPRs so the gather/scatter
// lower to SADDR-form global ops; each lane owns F/32 contiguous features.
// ---------------------------------------------------------------------------
template<int F>
__global__ void k_aggregate(const float* __restrict__ h,
                            const long long* __restrict__ src,
                            const long long* __restrict__ dst,
                            const float* __restrict__ dinv,
                            float* __restrict__ out) {
    const int lane = threadIdx.x & 31;
    const int e = blockIdx.x * (blockDim.x >> 5) + (threadIdx.x >> 5);
    if (e >= E_EDGES + N_NODES) return;

    int s, d;
    if (e < E_EDGES) {
        s = (int)src[e];
        d = (int)dst[e];
    } else {
        s = d = e - E_EDGES;   // self loop
    }
    // e is uniform per wave -> s, d are uniform; make that provable.
    s = __builtin_amdgcn_readfirstlane(s);
    d = __builtin_amdgcn_readfirstlane(d);
    const float norm = dinv[s] * dinv[d];

    const float* hs = h + (size_t)s * F;
    float* od = out + (size_t)d * F;

    if constexpr (F == 128) {
        const float4 v = *(const float4*)(hs + lane * 4);
        atomicAdd(&od[lane * 4 + 0], v.x * norm);
        atomicAdd(&od[lane * 4 + 1], v.y * norm);
        atomicAdd(&od[lane * 4 + 2], v.z * norm);
        atomicAdd(&od[lane * 4 + 3], v.w * norm);
    } else {
        const float2 v = *(const float2*)(hs + lane * 2);
        atomicAdd(&od[lane * 2 + 0], v.x * norm);
        atomicAdd(&od[lane * 2 + 1], v.y * norm);
    }
}

__global__ void k_bias_relu(float* __restrict__ a, const float* __restrict__ b) {
    int i = blockIdx.x * blockDim.x + threadIdx.x;
    if (i < N_NODES * F_HID) {
        float v = a[i] + b[i & (F_HID - 1)];
        a[i] = v > 0.0f ? v : 0.0f;
    }
}

__global__ void k_bias(float* __restrict__ a, const float* __restrict__ b) {
    int i = blockIdx.x * blockDim.x + threadIdx.x;
    if (i < N_NODES * F_OUT) a[i] += b[i & (F_OUT - 1)];
}

// ---------------------------------------------------------------------------
extern "C" void kernel_launch(void* const* d_in, const int* in_sizes, int n_in,
                              void* d_out, int out_size, void* d_ws, size_t ws_size,
                              hipStream_t stream) {
    const float*     x   = (const float*)d_in[0];        // [N, 128]
    const float*     W1  = (const float*)d_in[1];        // [128, 128]
    const float*     b1  = (const float*)d_in[2];        // [128]
    const float*     W2  = (const float*)d_in[3];        // [128, 64]
    const float*     b2  = (const float*)d_in[4];        // [64]
    const long long* ei  = (const long long*)d_in[5];    // [2, 800000] int64
    const long long* src = ei;
    const long long* dst = ei + E_EDGES;
    float* out = (float*)d_out;                          // [N, 64]

    // Workspace layout (floats); keep 16B alignment for vector loads.
    float* base = (float*)d_ws;
    float* deg  = base;                                  // N (-> dinv in place)
    float* W1t  = base + 50048;                          // 128*128
    float* W2t  = W1t  + F_IN * F_HID;                   // 128*64
    float* h1   = W2t  + F_HID * F_OUT;                  // N*128
    float* agg1 = h1   + (size_t)N_NODES * F_HID;        // N*128
    float* h2   = agg1 + (size_t)N_NODES * F_HID;        // N*64
    (void)ws_size; (void)in_sizes; (void)n_in; (void)out_size;

    const int T = 256;

    // 1) init accumulators + self-loop degrees; transpose weights
    k_init<<<(N_NODES * F_HID + T - 1) / T, T, 0, stream>>>(deg, agg1, out);
    k_transpose<F_IN,  F_HID><<<(F_IN * F_HID + T - 1) / T, T, 0, stream>>>(W1, W1t);
    k_transpose<F_HID, F_OUT><<<(F_HID * F_OUT + T - 1) / T, T, 0, stream>>>(W2, W2t);
    // 2) degree counts, 3) dinv
    k_degree<<<(E_EDGES + T - 1) / T, T, 0, stream>>>(dst, deg);
    k_rsqrt<<<(N_NODES + T - 1) / T, T, 0, stream>>>(deg);
    // 4) h1 = x @ W1   (3125 node tiles, 8 waves = 8 col tiles per block)
    k_gemm_wmma<F_IN, F_HID><<<N_NODES / 16, (F_HID / 16) * 32, 0, stream>>>(x, W1t, h1);
    // 5) agg1 += h1[src] * norm  (one wave per edge; 8 waves/block)
    {
        const int nE = E_EDGES + N_NODES;
        k_aggregate<F_HID><<<(nE + 7) / 8, 256, 0, stream>>>(h1, src, dst, deg, agg1);
    }
    // 6) agg1 = relu(agg1 + b1)
    k_bias_relu<<<(N_NODES * F_HID + T - 1) / T, T, 0, stream>>>(agg1, b1);
    // 7) h2 = agg1 @ W2  (4 col tiles -> 4 waves per block)
    k_gemm_wmma<F_HID, F_OUT><<<N_NODES / 16, (F_OUT / 16) * 32, 0, stream>>>(agg1, W2t, h2);
    // 8) out += h2[src] * norm
    {
        const int nE = E_EDGES + N_NODES;
        k_aggregate<F_OUT><<<(nE + 7) / 8, 256, 0, stream>>>(h2, src, dst, deg, out);
    }
    // 9) out += b2
    k_bias<<<(N_NODES * F_OUT + T - 1) / T, T, 0, stream>>>(out, b2);
}